// GAT_69535520522414
// MI455X (gfx1250) — compile-verified
//
#include <hip/hip_runtime.h>
#include <hip/hip_bf16.h>
#include <stdint.h>

typedef __attribute__((ext_vector_type(2))) float v2f;
typedef __attribute__((ext_vector_type(8))) float v8f;

#define LEAKY_SLOPE 0.2f

// ---------------------------------------------------------------------------
// generic fill (used to init accumulators / -inf max buffers every launch)
// ---------------------------------------------------------------------------
__global__ void gat_fill_u32(unsigned* __restrict__ p, unsigned v, long long n) {
  long long i = (long long)blockIdx.x * blockDim.x + threadIdx.x;
  if (i < n) p[i] = v;
}

// ---------------------------------------------------------------------------
// C[M,Ncol] = A[M,K] @ B[K,Ncol], all f32 row-major, via V_WMMA_F32_16X16X4_F32.
// One wave32 computes one 16x16 C tile. 8 waves per block share a 16xK A tile
// staged in LDS (row stride 132 floats -> conflict-free b64 fragment loads).
//
// A fragment (ISA 16x4 f32 layout): lanes 0-15 hold rows M=0..15 with
// K={k0,k0+1}; lanes 16-31 hold the same rows with K={k0+2,k0+3}.
// B fragment mirrored; C/D: VGPR i = row (i + 8*half), col = lane%16.
// ---------------------------------------------------------------------------
__global__ void gat_wmma_gemm_f32(const float* __restrict__ A,
                                  const float* __restrict__ B,
                                  float* __restrict__ C,
                                  int M, int K, int Ncol) {
  __shared__ float As[16 * 132];

  int tileM = blockIdx.x;
  // cooperative A-tile load (16 rows x K cols)
  for (int i = threadIdx.x; i < 16 * K; i += blockDim.x) {
    int r = i / K, c = i % K;
    int gr = tileM * 16 + r;
    As[r * 132 + c] = (gr < M) ? A[(size_t)gr * K + c] : 0.0f;
  }
  __syncthreads();

  int lane = threadIdx.x & 31;
  int wave = threadIdx.x >> 5;
  int colTile = blockIdx.y * (blockDim.x >> 5) + wave;
  if (colTile * 16 >= Ncol) return;          // wave-uniform: EXEC stays all-1s

  int half = lane >> 4;                      // 0: lanes 0-15, 1: lanes 16-31
  int l16  = lane & 15;
  int col  = colTile * 16 + l16;

  v8f acc = {};
  for (int k0 = 0; k0 < K; k0 += 4) {
    int ka = k0 + half * 2;
    v2f a, b;
    a.x = As[l16 * 132 + ka];
    a.y = As[l16 * 132 + ka + 1];
    b.x = B[(size_t)ka * Ncol + col];
    b.y = B[(size_t)(ka + 1) * Ncol + col];
    acc = __builtin_amdgcn_wmma_f32_16x16x4_f32(
        /*neg_a=*/false, a, /*neg_b=*/false, b,
        /*c_mod=*/(short)0, acc, /*reuse_a=*/false, /*reuse_b=*/false);
  }
#pragma unroll
  for (int i = 0; i < 8; ++i) {
    int r = tileM * 16 + half * 8 + i;
    if (r < M) C[(size_t)r * Ncol + col] = acc[i];
  }
}

// ---------------------------------------------------------------------------
// alpha_src[n,h] = sum_c h[n,h,c]*a_s[h,c] ; alpha_dst likewise
// ---------------------------------------------------------------------------
__global__ void gat_attn_coef(const float* __restrict__ h,
                              const float* __restrict__ a_s,
                              const float* __restrict__ a_d,
                              float* __restrict__ asrc,
                              float* __restrict__ adst,
                              int N, int H, int C) {
  int idx = blockIdx.x * blockDim.x + threadIdx.x;
  if (idx >= N * H) return;
  int n = idx / H, hh = idx % H;
  const float* hv = h + (size_t)n * H * C + (size_t)hh * C;
  float s = 0.f, d = 0.f;
  for (int c = 0; c < C; ++c) {
    float v = hv[c];
    s += v * a_s[hh * C + c];
    d += v * a_d[hh * C + c];
  }
  asrc[idx] = s;
  adst[idx] = d;
}

// edge endpoint lookup with implicit self-loops appended at eid in [E, E+N)
__device__ __forceinline__ void gat_edge_ends(const int* __restrict__ ei,
                                              int E, int eid, int& s, int& d) {
  if (eid < E) { s = ei[eid]; d = ei[E + eid]; }
  else         { s = eid - E; d = eid - E; }
}

__device__ __forceinline__ float gat_leaky(float e) {
  return e > 0.f ? e : LEAKY_SLOPE * e;
}

// float atomic-max via signed-max / unsigned-min trick (init = -inf bits)
__device__ __forceinline__ void gat_atomic_max_f32(unsigned* addr, float val) {
  if (val >= 0.f) atomicMax((int*)addr, __float_as_int(val));
  else            atomicMin(addr, __float_as_uint(val));
}

// pass 1: per-destination running max of leaky attention logits
__global__ void gat_edge_max(const int* __restrict__ ei, int E, int N, int H,
                             const float* __restrict__ asrc,
                             const float* __restrict__ adst,
                             unsigned* __restrict__ emax) {
  long long idx = (long long)blockIdx.x * blockDim.x + threadIdx.x;
  long long total = (long long)(E + N) * H;
  if (idx >= total) return;
  int eid = (int)(idx / H), hh = (int)(idx % H);
  int s, d; gat_edge_ends(ei, E, eid, s, d);
  float e = gat_leaky(asrc[(size_t)s * H + hh] + adst[(size_t)d * H + hh]);
  gat_atomic_max_f32(&emax[(size_t)d * H + hh], e);
}

// pass 2: ex = exp(e - max), cache ex, accumulate softmax denominator
__global__ void gat_edge_exp(const int* __restrict__ ei, int E, int N, int H,
                             const float* __restrict__ asrc,
                             const float* __restrict__ adst,
                             const unsigned* __restrict__ emax,
                             float* __restrict__ exbuf,
                             float* __restrict__ den) {
  long long idx = (long long)blockIdx.x * blockDim.x + threadIdx.x;
  long long total = (long long)(E + N) * H;
  if (idx >= total) return;
  int eid = (int)(idx / H), hh = (int)(idx % H);
  int s, d; gat_edge_ends(ei, E, eid, s, d);
  float e = gat_leaky(asrc[(size_t)s * H + hh] + adst[(size_t)d * H + hh]);
  float m = __uint_as_float(emax[(size_t)d * H + hh]);
  float ex = __expf(e - m);
  exbuf[idx] = ex;
  atomicAdd(&den[(size_t)d * H + hh], ex);
}

// pass 3: out[dst, h, c] += (ex/denom) * feat[src, h, c]
__global__ void gat_edge_aggregate(const int* __restrict__ ei, int E, int N,
                                   int H, int C,
                                   const float* __restrict__ exbuf,
                                   const float* __restrict__ den,
                                   const float* __restrict__ feat,
                                   float* __restrict__ out) {
  long long idx = (long long)blockIdx.x * blockDim.x + threadIdx.x;
  int HC = H * C;
  long long total = (long long)(E + N) * HC;
  if (idx >= total) return;
  int f   = (int)(idx % HC);
  int eid = (int)(idx / HC);
  int hh  = f / C;
  int s, d; gat_edge_ends(ei, E, eid, s, d);
  float alpha = exbuf[(size_t)eid * H + hh] /
                (den[(size_t)d * H + hh] + 1e-16f);
  atomicAdd(&out[(size_t)d * HC + f], alpha * feat[(size_t)s * HC + f]);
}

// h = elu(h + b), b broadcast over last dim F
__global__ void gat_elu_bias(float* __restrict__ h, const float* __restrict__ b,
                             int F, long long n) {
  long long i = (long long)blockIdx.x * blockDim.x + threadIdx.x;
  if (i >= n) return;
  float v = h[i] + b[(int)(i % F)];
  h[i] = v > 0.f ? v : __expf(v) - 1.0f;
}

// out += b, b broadcast over last dim F
__global__ void gat_add_bias(float* __restrict__ o, const float* __restrict__ b,
                             int F, long long n) {
  long long i = (long long)blockIdx.x * blockDim.x + threadIdx.x;
  if (i >= n) return;
  o[i] = o[i] + b[(int)(i % F)];
}

// ---------------------------------------------------------------------------
extern "C" void kernel_launch(void* const* d_in, const int* in_sizes, int n_in,
                              void* d_out, int out_size, void* d_ws, size_t ws_size,
                              hipStream_t stream) {
  const float* x   = (const float*)d_in[0];
  const int*   ei  = (const int*)  d_in[1];
  const float* W1  = (const float*)d_in[2];
  const float* as1 = (const float*)d_in[3];
  const float* ad1 = (const float*)d_in[4];
  const float* b1  = (const float*)d_in[5];
  const float* W2  = (const float*)d_in[6];
  const float* as2 = (const float*)d_in[7];
  const float* ad2 = (const float*)d_in[8];
  const float* b2  = (const float*)d_in[9];
  float* out = (float*)d_out;

  const int FIN = 128, H1 = 8, C1 = 16, HID = H1 * C1 /*128*/, CLS = 16;
  const int N  = in_sizes[0] / FIN;
  const int E  = in_sizes[1] / 2;
  const int E2 = E + N;

  // -------- workspace layout --------
  char* w = (char*)d_ws;
  auto alloc = [&](size_t bytes) -> char* {
    char* p = w;
    w += (bytes + 255) & ~(size_t)255;
    return p;
  };
  float*    h1    = (float*)   alloc((size_t)N * HID * 4);   // layer-1 features
  float*    h2    = (float*)   alloc((size_t)N * HID * 4);   // layer-1 output / layer-2 input
  float*    g     = (float*)   alloc((size_t)N * CLS * 4);   // layer-2 features
  float*    asr1  = (float*)   alloc((size_t)N * H1 * 4);
  float*    ads1  = (float*)   alloc((size_t)N * H1 * 4);
  unsigned* emax1 = (unsigned*)alloc((size_t)N * H1 * 4);
  float*    den1  = (float*)   alloc((size_t)N * H1 * 4);
  float*    asr2  = (float*)   alloc((size_t)N * 4);
  float*    ads2  = (float*)   alloc((size_t)N * 4);
  unsigned* emax2 = (unsigned*)alloc((size_t)N * 4);
  float*    den2  = (float*)   alloc((size_t)N * 4);
  float*    exbuf = (float*)   alloc((size_t)E2 * H1 * 4);   // softmax numerators

  const int BT = 256;
  auto blocks = [&](long long n) { return (unsigned)((n + BT - 1) / BT); };
  const unsigned NEG_INF = 0xFF800000u;

  // -------- init accumulators (every launch: graph-replay safe) --------
  gat_fill_u32<<<blocks((long long)N * H1), BT, 0, stream>>>(emax1, NEG_INF, (long long)N * H1);
  gat_fill_u32<<<blocks((long long)N * H1), BT, 0, stream>>>((unsigned*)den1, 0u, (long long)N * H1);
  gat_fill_u32<<<blocks((long long)N * HID), BT, 0, stream>>>((unsigned*)h2, 0u, (long long)N * HID);
  gat_fill_u32<<<blocks(N), BT, 0, stream>>>(emax2, NEG_INF, N);
  gat_fill_u32<<<blocks(N), BT, 0, stream>>>((unsigned*)den2, 0u, N);
  gat_fill_u32<<<blocks((long long)N * CLS), BT, 0, stream>>>((unsigned*)out, 0u, (long long)N * CLS);

  const int rowTiles = (N + 15) / 16;

  // ================= layer 1 =================
  {
    dim3 grid(rowTiles, (HID / 16 + 7) / 8);
    gat_wmma_gemm_f32<<<grid, BT, 0, stream>>>(x, W1, h1, N, FIN, HID);
  }
  gat_attn_coef<<<blocks((long long)N * H1), BT, 0, stream>>>(h1, as1, ad1, asr1, ads1, N, H1, C1);
  gat_edge_max<<<blocks((long long)E2 * H1), BT, 0, stream>>>(ei, E, N, H1, asr1, ads1, emax1);
  gat_edge_exp<<<blocks((long long)E2 * H1), BT, 0, stream>>>(ei, E, N, H1, asr1, ads1, emax1, exbuf, den1);
  gat_edge_aggregate<<<blocks((long long)E2 * HID), BT, 0, stream>>>(ei, E, N, H1, C1, exbuf, den1, h1, h2);
  gat_elu_bias<<<blocks((long long)N * HID), BT, 0, stream>>>(h2, b1, HID, (long long)N * HID);

  // ================= layer 2 (single head, mean == identity) =================
  {
    dim3 grid(rowTiles, 1);
    gat_wmma_gemm_f32<<<grid, BT, 0, stream>>>(h2, W2, g, N, HID, CLS);
  }
  gat_attn_coef<<<blocks(N), BT, 0, stream>>>(g, as2, ad2, asr2, ads2, N, 1, CLS);
  gat_edge_max<<<blocks(E2), BT, 0, stream>>>(ei, E, N, 1, asr2, ads2, emax2);
  gat_edge_exp<<<blocks(E2), BT, 0, stream>>>(ei, E, N, 1, asr2, ads2, emax2, exbuf, den2);
  gat_edge_aggregate<<<blocks((long long)E2 * CLS), BT, 0, stream>>>(ei, E, N, 1, CLS, exbuf, den2, g, out);
  gat_add_bias<<<blocks((long long)N * CLS), BT, 0, stream>>>(out, b2, CLS, (long long)N * CLS);
}